// Conv_RBS_state_vector_1632087572677
// MI455X (gfx1250) — compile-verified
//
#include <hip/hip_runtime.h>

// Conv_RBS_state_vector on MI455X (gfx1250):
//   1) extract Givens pairs from M1            (one pass over M1)
//   2) compose full-circuit unitary W = U^T    (LDS-resident, column-parallel)
//   3) out = in @ W via V_WMMA_F32_16X16X4_F32 (single 2048x496x496 f32 GEMM)

typedef __attribute__((ext_vector_type(2))) float v2f;
typedef __attribute__((ext_vector_type(8))) float v8f;

#define D        496   // Hamming-weight-2 basis size C(32,2)
#define LDW      512   // padded leading dim for W
#define PAIR_CAP 64    // >= 30 pairs per tuple
#define TUP_CAP  64    // >= 30 tuples

// ---------------------------------------------------------------------------
// Kernel A: for each (tuple t, row i) find partner j with M1[t][i][j] = +1.
// Each rotation pair appears once (the +1 row is the "q-side" row).
// ---------------------------------------------------------------------------
__global__ void extract_pairs_k(const float* __restrict__ M1, int T,
                                int* __restrict__ counts,
                                int2* __restrict__ pairs) {
  int idx = blockIdx.x * blockDim.x + threadIdx.x;   // t*D + i
  if (idx >= T * D) return;
  int t = idx / D, i = idx - t * D;
  const float* row = M1 + (size_t)t * D * D + (size_t)i * D;
  int j = -1; float sv = 0.f;
  for (int c = 0; c < D; ++c) {
    float v = row[c];
    if (v != 0.f) { j = c; sv = v; }
  }
  if (j >= 0 && sv > 0.f) {
    int slot = atomicAdd(&counts[t], 1);             // order-free: pairs disjoint
    pairs[t * PAIR_CAP + slot] = make_int2(i, j);
  }
}

// ---------------------------------------------------------------------------
// Kernel B: build W = U^T where U = product of all gate rotations.
// Each thread owns one U-column (length 496) held in LDS; gates are disjoint
// row-pair rotations, so the hot loop needs no synchronization at all.
// LDS: 496 rows x 129 floats (pad 129 kills bank conflicts in the transpose
// write-out). 4 blocks x 128 threads cover the 512 padded columns.
// ---------------------------------------------------------------------------
__global__ void build_w_k(const float* __restrict__ theta,
                          const int* __restrict__ gti,
                          const int* __restrict__ gpi,
                          const int* __restrict__ counts,
                          const int2* __restrict__ pairs,
                          float* __restrict__ W, int G) {
  extern __shared__ float Uls[];                     // [D][129]
  const int kl = threadIdx.x;                        // 0..127
  const int k0 = blockIdx.x * 128;
  const int k  = k0 + kl;                            // global U column

  for (int i = 0; i < D; ++i)
    Uls[i * 129 + kl] = (i == k) ? 1.f : 0.f;        // identity (cols >=496 stay 0)

  for (int g = 0; g < G; ++g) {
    int t = gti[g], p = gpi[g];                      // uniform -> scalar loads
    float th = theta[p];
    float c = cosf(th), s = sinf(th);
    int n = counts[t];
    const int2* pl = pairs + t * PAIR_CAP;
    for (int pp = 0; pp < n; ++pp) {                 // 30 disjoint row pairs
      int2 ij = pl[pp];
      float a = Uls[ij.x * 129 + kl];
      float b = Uls[ij.y * 129 + kl];
      Uls[ij.x * 129 + kl] =  c * a + s * b;
      Uls[ij.y * 129 + kl] = -s * a + c * b;
    }
  }

  __syncthreads();                                   // cross-column reads below
  // Write slice transposed: W[k][i] = U[i][k]; coalesced global stores.
  for (int kk = 0; kk < 128; ++kk) {
    int krow = k0 + kk;
    for (int ib = threadIdx.x; ib < LDW; ib += 128) {
      float v = (ib < D) ? Uls[ib * 129 + kk] : 0.f;
      W[(size_t)krow * LDW + ib] = v;
    }
  }
}

// ---------------------------------------------------------------------------
// Kernel C: out[b,n] = sum_k in[b,k] * W[k,n] via V_WMMA_F32_16X16X4_F32.
// Block = 8 waves; each wave computes a 64(M) x 16(N) tile: 4 accumulators,
// one B fragment shared across 4 WMMAs per K-step. K = 496 = 124 * 4.
// Fragment layouts per CDNA5 ISA 7.12.2 (32-bit A 16x4, B 4x16, C/D 16x16).
// ---------------------------------------------------------------------------
__global__ void gemm_wmma_k(const float* __restrict__ A,   // [B, 496]
                            const float* __restrict__ W,   // [512, 512] (K-major)
                            float* __restrict__ out) {     // [B, 496]
  const int lane = threadIdx.x & 31;
  const int wave = threadIdx.x >> 5;
  const int l16  = lane & 15;
  const int lh2  = (lane >> 4) * 2;                  // K sub-block base {0,2}
  const int m0   = (blockIdx.y * 8 + wave) * 64;
  const int n0   = blockIdx.x * 16;

  v8f acc[4];
#pragma unroll
  for (int mm = 0; mm < 4; ++mm) acc[mm] = (v8f)0.f;

  const float* a0 = A + (size_t)(m0 + l16) * D;      // A row for M = l16
  const float* bp = W + (size_t)n0 + l16;            // B col for N = l16

  for (int kk = 0; kk < D; kk += 4) {
    v2f b;                                           // B frag: K = kk+lh2(+1)
    b.x = bp[(size_t)(kk + lh2) * LDW];
    b.y = bp[(size_t)(kk + lh2 + 1) * LDW];
#pragma unroll
    for (int mm = 0; mm < 4; ++mm) {
      v2f a = *(const v2f*)(a0 + (size_t)mm * 16 * D + kk + lh2);  // 8B aligned
      acc[mm] = __builtin_amdgcn_wmma_f32_16x16x4_f32(
          false, a, false, b, (short)0, acc[mm], false, false);
    }
  }

  const int rbase = m0 + ((lane >> 4) << 3);         // C/D: M = 8*half + vgpr
#pragma unroll
  for (int mm = 0; mm < 4; ++mm)
#pragma unroll
    for (int v = 0; v < 8; ++v)
      out[(size_t)(rbase + mm * 16 + v) * D + n0 + l16] = acc[mm][v];
}

// ---------------------------------------------------------------------------
extern "C" void kernel_launch(void* const* d_in, const int* in_sizes, int n_in,
                              void* d_out, int out_size, void* d_ws, size_t ws_size,
                              hipStream_t stream) {
  const float* input_state = (const float*)d_in[0];  // [B, 496]
  const float* theta       = (const float*)d_in[1];  // [12]
  const float* M1          = (const float*)d_in[3];  // [T, 496, 496]
  const int*   gti         = (const int*)d_in[5];    // [G]
  const int*   gpi         = (const int*)d_in[6];    // [G]

  const int B = in_sizes[0] / D;                     // 2048
  const int T = in_sizes[2] / (D * D);               // 30
  const int G = in_sizes[5];                         // 156

  // Workspace layout: W (1 MB) | counts (TUP_CAP ints) | pairs (T*PAIR_CAP int2)
  float* W      = (float*)d_ws;
  int*   counts = (int*)((char*)d_ws + (size_t)LDW * LDW * sizeof(float));
  int2*  pairs  = (int2*)(counts + TUP_CAP);
  float* out    = (float*)d_out;

  hipMemsetAsync(counts, 0, TUP_CAP * sizeof(int), stream);

  int tot = T * D;
  extract_pairs_k<<<(tot + 255) / 256, 256, 0, stream>>>(M1, T, counts, pairs);

  build_w_k<<<4, 128, (size_t)D * 129 * sizeof(float), stream>>>(
      theta, gti, gpi, counts, pairs, W, G);

  dim3 grid(D / 16, B / 512);                        // 31 x 4
  gemm_wmma_k<<<grid, 256, 0, stream>>>(input_state, W, out);
}